// CTR_12232066859248
// MI455X (gfx1250) — compile-verified
//
#include <hip/hip_runtime.h>

// ---------------- problem constants ----------------
constexpr int   BN    = 4096;
constexpr int   KN    = 128;
constexpr float ALPHA = 20.0f;
constexpr float THR   = 0.005f;
constexpr float EPSV  = 1e-16f;

// ---------------- tiling constants -----------------
// KST=132: bank = (4*lane + k) % 64 for b64 column reads -> all 64 banks hit
// exactly once per wave (conflict-free), and 8-byte alignment everywhere.
constexpr int KST   = 132;
constexpr int WAVES = 4;
constexpr int BLOCK = WAVES * 32;     // 128 threads
constexpr int ROWS  = 16;             // batch rows per block = one WMMA M-tile
constexpr int GRID  = BN / ROWS;      // 256 blocks
constexpr int TPW   = 8 / WAVES;      // 16-col N-tiles per wave = 2

// workspace layout (floats):
//   [0]=err1 [1]=err2  [8..8+GRID) loss partials  [512..) u (BN*KN) then v (BN*KN)
constexpr int WS_PART = 8;
constexpr int WS_U    = 512;

typedef __attribute__((ext_vector_type(2))) float v2f;
typedef __attribute__((ext_vector_type(8))) float v8f;

// num/den via v_rcp_f32 + one Newton-Raphson step (~0.5 ulp, 4 VALU ops vs the
// ~12-op IEEE expansion). Sinkhorn is a fixed-point iteration; this does not
// disturb convergence at the 5e-3 threshold.
__device__ __forceinline__ float fast_div(float num, float den) {
  float r = __builtin_amdgcn_rcpf(den);
  r = __builtin_fmaf(__builtin_fmaf(-den, r, 1.0f), r, r);
  return num * r;
}

// Cooperative 16-row GEMM: the block's 4 waves split the 8 output N-tiles,
// each wave computing TPW tiles (t = t0..t0+TPW-1) of C = A(16xKN) @ B(KNxKN).
// A: block-shared LDS slab `aw` (stride KST).
// B: `bmT` stores B transposed (bmT[n*KST + k] = B[k][n]) so every B fragment
//    is one aligned, bank-conflict-free ds_load_b64.
// Even/odd K-chunks use separate accumulators -> 2*TPW independent WMMA chains.
__device__ __forceinline__ void wave_gemm(const float* aw, const float* bmT,
                                          int t0, int half, int l16,
                                          v8f out[TPW]) {
  const v8f vz = {0, 0, 0, 0, 0, 0, 0, 0};
  v8f acc[2][TPW];
#pragma unroll
  for (int p = 0; p < 2; ++p)
#pragma unroll
    for (int tt = 0; tt < TPW; ++tt) acc[p][tt] = vz;

#pragma unroll 4
  for (int c = 0; c < KN / 4; ++c) {
    const int k0 = 4 * c + 2 * half;                       // K pair for this half-wave
    const v2f af = *(const v2f*)(aw + l16 * KST + k0);     // A: M=l16, K=k0,k0+1
#pragma unroll
    for (int tt = 0; tt < TPW; ++tt) {
      const v2f bf = *(const v2f*)(bmT + (16 * (t0 + tt) + l16) * KST + k0);
      acc[c & 1][tt] = __builtin_amdgcn_wmma_f32_16x16x4_f32(
          false, af, false, bf, (short)0, acc[c & 1][tt], false, false);
    }
  }
#pragma unroll
  for (int tt = 0; tt < TPW; ++tt) out[tt] = acc[0][tt] + acc[1][tt];
}

__global__ void init_flags(float* ws) {
  if (threadIdx.x < 2) ws[threadIdx.x] = 0.0f;
}

// Runs `niter` Sinkhorn iterations fully on-chip for a 16-row slab.
// Optionally writes the marginal error to ws[err_slot]; optionally gated on
// earlier err flags (skip phase if err <= THR, matching the reference loop).
__global__ __launch_bounds__(BLOCK) void sinkhorn_phase(
    const float* __restrict__ ga, const float* __restrict__ gb,
    const float* __restrict__ gM, float* __restrict__ ws,
    int niter, int err_slot, int gate0, int gate1, int load_u) {
  if (gate0 >= 0 && !(ws[gate0] > THR)) return;   // uniform per block
  if (gate1 >= 0 && !(ws[gate1] > THR)) return;

  __shared__ __align__(16) float kmat [KN * KST];  // kmat [i*KST+j] = K[i][j]  (GEMM2: B = K^T)
  __shared__ __align__(16) float kmatT[KN * KST];  // kmatT[j*KST+i] = K[i][j]  (GEMM1: B = K)
  __shared__ __align__(16) float uscr[ROWS * KST];
  __shared__ __align__(16) float vscr[ROWS * KST];
  __shared__ float errp[WAVES * ROWS];

  const int tid  = threadIdx.x;
  const int wave = tid >> 5, lane = tid & 31;
  const int half = lane >> 4, l16 = lane & 15;
  const int t0   = wave * TPW;
  const int rowBase = blockIdx.x * ROWS;

  float* gu = ws + WS_U;
  float* gv = gu + BN * KN;

  // K = exp(-ALPHA*M), stored in both orientations
  for (int i = tid; i < KN * KN; i += BLOCK) {
    const int r = i >> 7, cc = i & 127;
    const float e = expf(-ALPHA * gM[i]);
    kmat [r  * KST + cc] = e;
    kmatT[cc * KST + r ] = e;
  }
  if (load_u) {
    for (int i = tid; i < ROWS * KN; i += BLOCK) {
      const int m = i >> 7, n = i & 127;
      uscr[m * KST + n] = gu[(rowBase + m) * KN + n];
    }
  } else {
    for (int i = tid; i < ROWS * KN; i += BLOCK) {
      const int m = i >> 7, n = i & 127;
      uscr[m * KST + n] = 1.0f / (float)KN;
    }
  }
  __syncthreads();

  for (int it = 0; it < niter; ++it) {
    // KTu = u @ K   (this wave's 2 N-tiles)
    v8f ktu[TPW];
    wave_gemm(uscr, kmatT, t0, half, l16, ktu);
    // v = b / (KTu + eps) for this wave's columns
#pragma unroll
    for (int tt = 0; tt < TPW; ++tt)
#pragma unroll
      for (int r = 0; r < 8; ++r) {
        const int m = r + 8 * half, n = 16 * (t0 + tt) + l16;
        vscr[m * KST + n] = fast_div(gb[(rowBase + m) * KN + n], ktu[tt][r] + EPSV);
      }
    __syncthreads();   // full v slab visible to all waves

    // Kv = v @ K^T  (this wave's 2 K-tiles)
    v8f kv[TPW];
    wave_gemm(vscr, kmat, t0, half, l16, kv);
    // u = a / (Kv + eps) for this wave's columns
#pragma unroll
    for (int tt = 0; tt < TPW; ++tt)
#pragma unroll
      for (int r = 0; r < 8; ++r) {
        const int m = r + 8 * half, n = 16 * (t0 + tt) + l16;
        uscr[m * KST + n] = fast_div(ga[(rowBase + m) * KN + n], kv[tt][r] + EPSV);
      }
    __syncthreads();   // full u slab visible for next iteration / err check
  }

  if (err_slot >= 0) {
    // err = max_b sum_j | v * (u_new @ K) - b |
    v8f bb[TPW];
    wave_gemm(uscr, kmatT, t0, half, l16, bb);
    float rs[8];
#pragma unroll
    for (int r = 0; r < 8; ++r) rs[r] = 0.0f;
#pragma unroll
    for (int tt = 0; tt < TPW; ++tt)
#pragma unroll
      for (int r = 0; r < 8; ++r) {
        const int m = r + 8 * half, n = 16 * (t0 + tt) + l16;
        rs[r] += fabsf(vscr[m * KST + n] * bb[tt][r] - gb[(rowBase + m) * KN + n]);
      }
#pragma unroll
    for (int r = 0; r < 8; ++r) {
      float s = rs[r];
      s += __shfl_xor(s, 1, 32);
      s += __shfl_xor(s, 2, 32);
      s += __shfl_xor(s, 4, 32);
      s += __shfl_xor(s, 8, 32);          // row sum over this wave's 32 columns
      if (l16 == 0) errp[wave * ROWS + 8 * half + r] = s;
    }
    __syncthreads();
    if (tid < ROWS) {                      // lanes 0..15 of wave 0
      float s = errp[tid] + errp[ROWS + tid] + errp[2 * ROWS + tid] + errp[3 * ROWS + tid];
      s = fmaxf(s, __shfl_xor(s, 1, 32));
      s = fmaxf(s, __shfl_xor(s, 2, 32));
      s = fmaxf(s, __shfl_xor(s, 4, 32));
      s = fmaxf(s, __shfl_xor(s, 8, 32));  // max over the 16 rows
      if (tid == 0)
        atomicMax(reinterpret_cast<int*>(ws + err_slot), __float_as_int(s));
    }
  }

  // persist u, v for the next phase / loss kernel
  for (int i = tid; i < ROWS * KN; i += BLOCK) {
    const int m = i >> 7, n = i & 127;
    gu[(rowBase + m) * KN + n] = uscr[m * KST + n];
    gv[(rowBase + m) * KN + n] = vscr[m * KST + n];
  }
}

// per-block partial of sum_b u_b^T (K .* M) v_b
__global__ __launch_bounds__(BLOCK) void loss_partial_k(
    const float* __restrict__ gM, float* __restrict__ ws) {
  __shared__ __align__(16) float wT[KN * KST];   // wT[j*KST+i] = W[i][j], W = K .* M
  __shared__ __align__(16) float uscr[ROWS * KST];
  __shared__ float wred[WAVES];

  const int tid  = threadIdx.x;
  const int wave = tid >> 5, lane = tid & 31;
  const int half = lane >> 4, l16 = lane & 15;
  const int t0   = wave * TPW;
  const int rowBase = blockIdx.x * ROWS;

  const float* gu = ws + WS_U;
  const float* gv = gu + BN * KN;

  for (int i = tid; i < KN * KN; i += BLOCK) {
    const int r = i >> 7, cc = i & 127;
    const float mv = gM[i];
    wT[cc * KST + r] = expf(-ALPHA * mv) * mv;
  }
  for (int i = tid; i < ROWS * KN; i += BLOCK) {
    const int m = i >> 7, n = i & 127;
    uscr[m * KST + n] = gu[(rowBase + m) * KN + n];
  }
  __syncthreads();

  v8f acc[TPW];
  wave_gemm(uscr, wT, t0, half, l16, acc);

  float part = 0.0f;
#pragma unroll
  for (int tt = 0; tt < TPW; ++tt)
#pragma unroll
    for (int r = 0; r < 8; ++r) {
      const int m = r + 8 * half, n = 16 * (t0 + tt) + l16;
      part += acc[tt][r] * gv[(rowBase + m) * KN + n];
    }
  part += __shfl_xor(part, 1, 32);
  part += __shfl_xor(part, 2, 32);
  part += __shfl_xor(part, 4, 32);
  part += __shfl_xor(part, 8, 32);
  part += __shfl_xor(part, 16, 32);
  if (lane == 0) wred[wave] = part;
  __syncthreads();
  if (tid == 0) {
    float s = 0.0f;
#pragma unroll
    for (int w = 0; w < WAVES; ++w) s += wred[w];
    ws[WS_PART + blockIdx.x] = s;   // fixed order -> deterministic
  }
}

__global__ void loss_final_k(const float* __restrict__ ws, float* __restrict__ out) {
  if (threadIdx.x == 0) {
    float s = 0.0f;
    for (int i = 0; i < GRID; ++i) s += ws[WS_PART + i];
    out[0] = s / (float)BN;   // WEIGHT_LOSS_CTR == 1.0
  }
}

extern "C" void kernel_launch(void* const* d_in, const int* in_sizes, int n_in,
                              void* d_out, int out_size, void* d_ws, size_t ws_size,
                              hipStream_t stream) {
  (void)in_sizes; (void)n_in; (void)out_size; (void)ws_size;
  const float* ga = (const float*)d_in[0];
  const float* gb = (const float*)d_in[1];
  const float* gM = (const float*)d_in[2];
  float* out = (float*)d_out;
  float* ws  = (float*)d_ws;

  init_flags<<<1, 32, 0, stream>>>(ws);
  // iter 1, err -> ws[0]
  sinkhorn_phase<<<GRID, BLOCK, 0, stream>>>(ga, gb, gM, ws, 1, 0, -1, -1, 0);
  // iters 2..51 (runs only if err1 > THR), err -> ws[1]
  sinkhorn_phase<<<GRID, BLOCK, 0, stream>>>(ga, gb, gM, ws, 50, 1, 0, -1, 1);
  // iters 52..100 (runs only if err1 > THR and err2 > THR)
  sinkhorn_phase<<<GRID, BLOCK, 0, stream>>>(ga, gb, gM, ws, 49, -1, 0, 1, 1);
  // loss = mean_b u^T (K .* M) v
  loss_partial_k<<<GRID, BLOCK, 0, stream>>>(gM, ws);
  loss_final_k<<<1, 32, 0, stream>>>(ws, out);
}